// ContrastiveLoss_44796508897758
// MI455X (gfx1250) — compile-verified
//
#include <hip/hip_runtime.h>

typedef __attribute__((ext_vector_type(2))) float v2f;
typedef __attribute__((ext_vector_type(8))) float v8f;

#define NROWS 4096
#define DIMK  512
#define BHALF 2048
#define INV_T (1.0f / 0.07f)
#define LDS_STRIDE 516   // 512 + 4 floats pad -> per-row bank offset of 4, no conflicts

// ---------------------------------------------------------------------------
// Kernel 0: zero the scalar output (d_out is poisoned by the harness).
// ---------------------------------------------------------------------------
__global__ void ntxent_zero_out(float* __restrict__ out) {
  if (threadIdx.x == 0) out[0] = 0.0f;
}

// ---------------------------------------------------------------------------
// Kernel 1: L2-normalize each row of features into workspace fn[4096*512].
// One block (128 threads = 4 waves) per row; each thread covers 4 columns.
// ---------------------------------------------------------------------------
__global__ __launch_bounds__(128) void ntxent_normalize(
    const float* __restrict__ f, float* __restrict__ fn) {
  const int row = blockIdx.x;
  const float* src = f + (size_t)row * DIMK;

  float vals[4];
  float ss = 0.0f;
#pragma unroll
  for (int t = 0; t < 4; ++t) {
    float v = src[threadIdx.x + t * 128];
    vals[t] = v;
    ss += v * v;
  }
  // wave32 reduction
#pragma unroll
  for (int off = 16; off >= 1; off >>= 1) ss += __shfl_xor(ss, off, 32);

  __shared__ float tmp[4];
  const int wid = threadIdx.x >> 5;
  if ((threadIdx.x & 31) == 0) tmp[wid] = ss;
  __syncthreads();
  ss = tmp[0] + tmp[1] + tmp[2] + tmp[3];

  const float inv = rsqrtf(ss);
#pragma unroll
  for (int t = 0; t < 4; ++t)
    fn[(size_t)row * DIMK + threadIdx.x + t * 128] = vals[t] * inv;
}

// ---------------------------------------------------------------------------
// Kernel 2: fused sim-GEMM (fp32 WMMA 16x16x4) + masked logsumexp + loss.
// Block = 128 threads (4 waves), one 16-row strip per block (256 blocks).
// Each wave sweeps 32 *pairs* of column tiles: one A-frag (LDS) feeds two
// WMMAs into two independent accumulators -> half the ds traffic per WMMA and
// two interleaved accumulation chains. The 16x16 sim tiles live only in
// accumulator VGPRs and are consumed immediately (64 MB sim never exists).
// Masked entries occur only in tiles with (i0^j0)&2047 == 0 (2 of 256), so
// the mask test is a wave-uniform scalar branch; the hot path is mul/exp/add.
// ---------------------------------------------------------------------------
__global__ __launch_bounds__(128) void ntxent_loss(
    const float* __restrict__ fn, float* __restrict__ out) {
  __shared__ float lds_a[16 * LDS_STRIDE];
  __shared__ float lds_neg[16];
  __shared__ float lds_pos[16];

  const int tid = threadIdx.x;
  const int i0  = blockIdx.x * 16;

  // Stage the 16 x 512 A-strip (normalized rows) into LDS, padded stride.
  for (int idx = tid; idx < 16 * DIMK; idx += 128) {
    const int r  = idx >> 9;          // /512
    const int cc = idx & (DIMK - 1);  // %512
    lds_a[r * LDS_STRIDE + cc] = fn[(size_t)(i0 + r) * DIMK + cc];
  }
  if (tid < 16) { lds_neg[tid] = 0.0f; lds_pos[tid] = 0.0f; }
  __syncthreads();

  const int lane = tid & 31;
  const int wave = tid >> 5;
  const int half = lane >> 4;   // 0: rows 0-7 / K 0-1 ; 1: rows 8-15 / K 2-3
  const int nl   = lane & 15;   // column-in-tile (B/C/D), row-in-tile (A)
  const int koff = half * 2;

  float negsum[8], possum[8];
#pragma unroll
  for (int r = 0; r < 8; ++r) { negsum[r] = 0.0f; possum[r] = 0.0f; }

  const float* aptr = &lds_a[nl * LDS_STRIDE + koff];

  for (int t = 0; t < 32; ++t) {
    const int j0 = (t * 4 + wave) * 32;           // first tile of the pair
    const float* bptr = fn + (size_t)(j0 + nl) * DIMK + koff;

    v8f c0 = {};
    v8f c1 = {};
#pragma unroll 8
    for (int k = 0; k < DIMK; k += 4) {
      // A 16x4 f32 frag (shared by both tiles): lane holds A[nl][koff..koff+1]
      v2f a  = *(const v2f*)(aptr + k);
      // B 4x16 f32 frags: lane holds B[koff..koff+1][nl] = fn[j][k+koff..]
      v2f b0 = *(const v2f*)(bptr + k);
      v2f b1 = *(const v2f*)(bptr + k + 16 * DIMK);   // +32KB immediate offset
      c0 = __builtin_amdgcn_wmma_f32_16x16x4_f32(false, a, false, b0,
                                                 (short)0, c0, false, false);
      c1 = __builtin_amdgcn_wmma_f32_16x16x4_f32(false, a, false, b1,
                                                 (short)0, c1, false, false);
    }

    // Consume both 16x16 sim tiles: lane owns col nl, rows half*8 + r.
#pragma unroll
    for (int p = 0; p < 2; ++p) {
      const int jt = j0 + p * 16;
      const int j  = jt + nl;
      const v8f& c = p ? c1 : c0;
      if (((i0 ^ jt) & (BHALF - 1)) == 0) {
        // Special tile (2 per strip): contains diagonal or positive entries.
        // Branchless per-element selects; positive picked up as a sum with
        // exactly one nonzero contribution block-wide.
#pragma unroll
        for (int r = 0; r < 8; ++r) {
          const int i = i0 + half * 8 + r;
          const int d = i ^ j;
          const float s = c[r];
          negsum[r] += ((d & (BHALF - 1)) == 0) ? 0.0f : __expf(s * INV_T);
          possum[r] += (d == BHALF) ? s : 0.0f;
        }
      } else {
        // Clean tile: pure exp-accumulate.
#pragma unroll
        for (int r = 0; r < 8; ++r) negsum[r] += __expf(c[r] * INV_T);
      }
    }
  }

  // Reduce across the 16 lanes that share each row (xor masks stay within the
  // half-wave), then combine the 4 waves via LDS atomics.
#pragma unroll
  for (int r = 0; r < 8; ++r) {
    float nv = negsum[r];
    float pv = possum[r];
#pragma unroll
    for (int off = 8; off >= 1; off >>= 1) {
      nv += __shfl_xor(nv, off, 32);
      pv += __shfl_xor(pv, off, 32);
    }
    if (nl == 0) {
      atomicAdd(&lds_neg[half * 8 + r], nv);
      atomicAdd(&lds_pos[half * 8 + r], pv);
    }
  }
  __syncthreads();

  // Per-row loss -> scalar mean via one global atomic per row.
  if (tid < 16) {
    const float rowloss = -lds_pos[tid] * INV_T + __logf(lds_neg[tid]);
    atomicAdd(out, rowloss * (1.0f / (float)NROWS));
  }
}

// ---------------------------------------------------------------------------
extern "C" void kernel_launch(void* const* d_in, const int* in_sizes, int n_in,
                              void* d_out, int out_size, void* d_ws, size_t ws_size,
                              hipStream_t stream) {
  (void)in_sizes; (void)n_in; (void)out_size; (void)ws_size;
  const float* features = (const float*)d_in[0];
  float* out = (float*)d_out;
  float* fn  = (float*)d_ws;   // 4096*512 floats = 8 MB of normalized features

  ntxent_zero_out<<<1, 32, 0, stream>>>(out);
  ntxent_normalize<<<NROWS, 128, 0, stream>>>(features, fn);
  ntxent_loss<<<NROWS / 16, 128, 0, stream>>>(fn, out);
}